// GATDecoder_69303592288577
// MI455X (gfx1250) — compile-verified
//
#include <hip/hip_runtime.h>
#include <hip/hip_bf16.h>
#include <math.h>

typedef __attribute__((ext_vector_type(2))) float v2f;
typedef __attribute__((ext_vector_type(8))) float v8f;

// ---------------------------------------------------------------------------
// Order-preserving float <-> uint encoding for atomic max on floats.
// ---------------------------------------------------------------------------
__device__ __forceinline__ unsigned encF(float f) {
    unsigned u = __float_as_uint(f);
    return (u & 0x80000000u) ? ~u : (u | 0x80000000u);
}
__device__ __forceinline__ float decF(unsigned e) {
    unsigned u = (e & 0x80000000u) ? (e & 0x7FFFFFFFu) : ~e;
    return __uint_as_float(u);
}

// edge e in [0, E+N): first E are edge_index columns, last N are self loops
__device__ __forceinline__ void edge_sd(const int* __restrict__ ei, int E, int e,
                                        int& s, int& d) {
    if (e < E) { s = ei[e]; d = ei[E + e]; }
    else       { s = e - E; d = e - E; }
}

// ---------------------------------------------------------------------------
// fp32 WMMA GEMM: C[M,Nc] = X[M,K] @ W[K,Nc] (+ bias).
// One wave computes a 16x64 output strip: 4 accumulators share each A operand,
// so per k-step we issue 4 V_WMMA_F32_16X16X4_F32 with one A fetch.
// Requirements: M%16==0, Nc%64==0, K%4==0 (true for all shapes here).
// ---------------------------------------------------------------------------
__global__ void gemm_wmma_f32(const float* __restrict__ X,
                              const float* __restrict__ W,
                              const float* __restrict__ bias,
                              float* __restrict__ C,
                              int M, int K, int Nc) {
    int wave = (int)((blockIdx.x * blockDim.x + threadIdx.x) >> 5);
    int lane = (int)(threadIdx.x & 31u);
    int stripsN = Nc >> 6;                       // 64-wide strips
    int tilesM  = M >> 4;
    if (wave >= tilesM * stripsN) return;        // wave-uniform: EXEC stays full
    int tm = wave / stripsN;
    int tn = wave - tm * stripsN;

    int rc = lane & 15;        // A-row (lanes 0..15 twice) / B-col within tile
    int hi = lane >> 4;        // 0: K pair {0,1}; 1: K pair {2,3}

    const float* Xr = X + (size_t)(tm * 16 + rc) * K;
    const float* Wb = W + (tn * 64 + rc);

    v8f acc0 = {}, acc1 = {}, acc2 = {}, acc3 = {};
    for (int k = 0; k < K; k += 4) {
        int ka = k + 2 * hi;
        v2f a;
        a.x = Xr[ka];
        a.y = Xr[ka + 1];
        const float* w0 = Wb + (size_t)ka * Nc;
        const float* w1 = Wb + (size_t)(ka + 1) * Nc;
        v2f b0, b1, b2, b3;
        b0.x = w0[0];  b0.y = w1[0];
        b1.x = w0[16]; b1.y = w1[16];
        b2.x = w0[32]; b2.y = w1[32];
        b3.x = w0[48]; b3.y = w1[48];
        acc0 = __builtin_amdgcn_wmma_f32_16x16x4_f32(false, a, false, b0, (short)0, acc0, false, false);
        acc1 = __builtin_amdgcn_wmma_f32_16x16x4_f32(false, a, false, b1, (short)0, acc1, false, false);
        acc2 = __builtin_amdgcn_wmma_f32_16x16x4_f32(false, a, false, b2, (short)0, acc2, false, false);
        acc3 = __builtin_amdgcn_wmma_f32_16x16x4_f32(false, a, false, b3, (short)0, acc3, false, false);
    }

    // store: C/D layout -> VGPR r holds M = 16*tm + r + 8*hi, N = strip + j*16 + rc
    v8f accs[4] = {acc0, acc1, acc2, acc3};
#pragma unroll
    for (int j = 0; j < 4; ++j) {
        int n = tn * 64 + j * 16 + rc;
        float bv = bias ? bias[n] : 0.0f;
#pragma unroll
        for (int r = 0; r < 8; ++r) {
            int m = tm * 16 + r + 8 * hi;
            C[(size_t)m * Nc + n] = accs[j][r] + bv;
        }
    }
}

// ---------------------------------------------------------------------------
// Per-node attention logits: sA[n] = h[n,:]·a_src, dA[n] = h[n,:]·a_dst
// ---------------------------------------------------------------------------
__global__ void node_alpha(const float* __restrict__ H,
                           const float* __restrict__ as_, const float* __restrict__ ad_,
                           float* __restrict__ sA, float* __restrict__ dA,
                           int Nn, int Do) {
    int n = blockIdx.x * blockDim.x + threadIdx.x;
    if (n >= Nn) return;
    const float* h = H + (size_t)n * Do;
    float ss = 0.f, dd = 0.f;
    for (int c = 0; c < Do; ++c) {
        float v = h[c];
        ss = fmaf(v, as_[c], ss);
        dd = fmaf(v, ad_[c], dd);
    }
    sA[n] = ss;
    dA[n] = dd;
}

// zero accumulator + reset segment-max / denom
__global__ void init_layer(unsigned* __restrict__ amax, float* __restrict__ denom,
                           float* __restrict__ outbuf, int Nn, int Do) {
    size_t i = (size_t)blockIdx.x * blockDim.x + threadIdx.x;
    size_t tot = (size_t)Nn * Do;
    if (i < tot) outbuf[i] = 0.0f;
    if (i < (size_t)Nn) { amax[i] = 0u; denom[i] = 0.0f; }
}

// pass 1: alpha = leaky_relu(sA[src]+dA[dst], 0.2); segment max over dst
__global__ void edge_alpha_max(const int* __restrict__ ei, int E, int Etot,
                               const float* __restrict__ sA, const float* __restrict__ dA,
                               float* __restrict__ abuf, unsigned* __restrict__ amax) {
    int e = blockIdx.x * blockDim.x + threadIdx.x;
    if (e >= Etot) return;
    int s, d; edge_sd(ei, E, e, s, d);
    float a = sA[s] + dA[d];
    a = (a > 0.0f) ? a : 0.2f * a;
    abuf[e] = a;
    atomicMax(&amax[d], encF(a));
}

// pass 2: ex = exp(alpha - amax[dst]); denom[dst] += ex
__global__ void edge_exp_sum(const int* __restrict__ ei, int E, int Etot,
                             float* __restrict__ abuf, const unsigned* __restrict__ amax,
                             float* __restrict__ denom) {
    int e = blockIdx.x * blockDim.x + threadIdx.x;
    if (e >= Etot) return;
    int s, d; edge_sd(ei, E, e, s, d);
    float ex = expf(abuf[e] - decF(amax[d]));
    abuf[e] = ex;
    atomicAdd(&denom[d], ex);
}

// pass 3: one wave per edge: out[dst,:] += h[src,:] * (ex / (denom[dst]+eps))
__global__ void edge_aggregate(const int* __restrict__ ei, int E, int Etot,
                               const float* __restrict__ abuf, const float* __restrict__ denom,
                               const float* __restrict__ H, float* __restrict__ Out, int Do) {
    int wave = (int)((blockIdx.x * blockDim.x + threadIdx.x) >> 5);
    int lane = (int)(threadIdx.x & 31u);
    if (wave >= Etot) return;                       // wave-uniform
    int s, d; edge_sd(ei, E, wave, s, d);
    float coef = abuf[wave] / (denom[d] + 1e-16f);
    const float* hs = H + (size_t)s * Do;
    float* od = Out + (size_t)d * Do;
    for (int c = lane; c < Do; c += 32)
        atomicAdd(&od[c], hs[c] * coef);
}

// out = leaky_relu(out + b, 0.01)
__global__ void bias_lrelu(float* __restrict__ Out, const float* __restrict__ b,
                           int Nn, int Do) {
    size_t i = (size_t)blockIdx.x * blockDim.x + threadIdx.x;
    if (i >= (size_t)Nn * Do) return;
    int c = (int)(i % (size_t)Do);
    float v = Out[i] + b[c];
    Out[i] = (v > 0.0f) ? v : 0.01f * v;
}

// inner[e] = sigmoid(aprim[src]·aprim[dst]), dim 64, one wave per edge
__global__ void edge_inner(const int* __restrict__ ei, int E,
                           const float* __restrict__ aprim, float* __restrict__ out) {
    int wave = (int)((blockIdx.x * blockDim.x + threadIdx.x) >> 5);
    int lane = (int)(threadIdx.x & 31u);
    if (wave >= E) return;                          // wave-uniform
    int s = ei[wave], d = ei[E + wave];
    const float* as_ = aprim + (size_t)s * 64;
    const float* ad_ = aprim + (size_t)d * 64;
    float acc = as_[lane] * ad_[lane] + as_[lane + 32] * ad_[lane + 32];
#pragma unroll
    for (int off = 16; off > 0; off >>= 1)
        acc += __shfl_xor(acc, off, 32);
    if (lane == 0) out[wave] = 1.0f / (1.0f + expf(-acc));
}

// ---------------------------------------------------------------------------
static inline int cdiv(long long a, long long b) { return (int)((a + b - 1) / b); }

extern "C" void kernel_launch(void* const* d_in, const int* in_sizes, int n_in,
                              void* d_out, int out_size, void* d_ws, size_t ws_size,
                              hipStream_t stream) {
    const float* z  = (const float*)d_in[0];
    const int*   ei = (const int*)d_in[1];

    const float* W[4]  = {(const float*)d_in[2],  (const float*)d_in[6],
                          (const float*)d_in[10], (const float*)d_in[14]};
    const float* As[4] = {(const float*)d_in[3],  (const float*)d_in[7],
                          (const float*)d_in[11], (const float*)d_in[15]};
    const float* Ad[4] = {(const float*)d_in[4],  (const float*)d_in[8],
                          (const float*)d_in[12], (const float*)d_in[16]};
    const float* B[4]  = {(const float*)d_in[5],  (const float*)d_in[9],
                          (const float*)d_in[13], (const float*)d_in[17]};
    const float* Wz = (const float*)d_in[18];
    const float* bz = (const float*)d_in[19];
    const float* Wa = (const float*)d_in[20];
    const float* ba = (const float*)d_in[21];

    const int Nn   = in_sizes[0] / 128;   // 50000
    const int Ee   = in_sizes[1] / 2;     // 800000
    const int Etot = Ee + Nn;             // + self loops

    const int Di[4] = {128, 64, 128, 192};
    const int Do[4] = {64, 128, 192, 256};

    // workspace layout (floats)
    float* ws = (float*)d_ws;
    size_t feat = (size_t)Nn * 256;
    float*    hbuf  = ws;                       // [N,256] max
    float*    buf1  = hbuf + feat;              // [N,256]
    float*    buf2  = buf1 + feat;              // [N,256]
    float*    sA    = buf2 + feat;              // [N]
    float*    dA    = sA + Nn;                  // [N]
    unsigned* amax  = (unsigned*)(dA + Nn);     // [N]
    float*    denom = (float*)(amax + Nn);      // [N]
    float*    abuf  = denom + Nn;               // [Etot]
    (void)ws_size; (void)n_in; (void)out_size;

    const float* X = z;
    float* obufs[4] = {buf1, buf2, buf1, buf2};

    for (int l = 0; l < 4; ++l) {
        int di = Di[l], dw = Do[l];
        float* Ob = obufs[l];

        // h = X @ W_l   (fp32 WMMA, 16x64 strip per wave)
        {
            long long waves = (long long)(Nn / 16) * (dw / 64);
            gemm_wmma_f32<<<cdiv(waves * 32, 128), 128, 0, stream>>>(
                X, W[l], nullptr, hbuf, Nn, di, dw);
        }
        // per-node logits
        node_alpha<<<cdiv(Nn, 256), 256, 0, stream>>>(hbuf, As[l], Ad[l], sA, dA, Nn, dw);
        // reset accumulators
        init_layer<<<cdiv((long long)Nn * dw, 256), 256, 0, stream>>>(amax, denom, Ob, Nn, dw);
        // segment softmax over dst
        edge_alpha_max<<<cdiv(Etot, 256), 256, 0, stream>>>(ei, Ee, Etot, sA, dA, abuf, amax);
        edge_exp_sum<<<cdiv(Etot, 256), 256, 0, stream>>>(ei, Ee, Etot, abuf, amax, denom);
        // weighted scatter: one wave per edge
        edge_aggregate<<<cdiv((long long)Etot * 32, 256), 256, 0, stream>>>(
            ei, Ee, Etot, abuf, denom, hbuf, Ob, dw);
        // bias + leaky_relu(0.01)
        bias_lrelu<<<cdiv((long long)Nn * dw, 256), 256, 0, stream>>>(Ob, B[l], Nn, dw);

        X = Ob;
    }

    float* zprim = (float*)d_out;              // [N,128]
    float* inner = zprim + (size_t)Nn * 128;   // [E]
    float* aprim = buf1;                       // [N,64] (buf1 free after layer 4)

    // zprim = h @ Wz + bz  -> directly to d_out
    {
        long long waves = (long long)(Nn / 16) * (128 / 64);
        gemm_wmma_f32<<<cdiv(waves * 32, 128), 128, 0, stream>>>(
            X, Wz, bz, zprim, Nn, 256, 128);
    }
    // aprim = h @ Wa + ba
    {
        long long waves = (long long)(Nn / 16) * (64 / 64);
        gemm_wmma_f32<<<cdiv(waves * 32, 128), 128, 0, stream>>>(
            X, Wa, ba, aprim, Nn, 256, 64);
    }
    // inner = sigmoid(aprim[src]·aprim[dst]) over original E edges
    edge_inner<<<cdiv((long long)Ee * 32, 256), 256, 0, stream>>>(ei, Ee, aprim, inner);
}